// DualTriangleAttention_28157805593364
// MI455X (gfx1250) — compile-verified
//
#include <hip/hip_runtime.h>
#include <hip/hip_bf16.h>
#include <cstdint>

typedef __attribute__((ext_vector_type(2))) float v2f;
typedef __attribute__((ext_vector_type(8))) float v8f;

// Problem constants (fixed by the reference)
#define BB 2
#define SS 2048
#define DD 1024
#define HH 16
#define DH 64
#define LDQKV 3072

#define APAD 36   // A-tile LDS row stride: 144B rows (16B-aligned b128 writes), even cols 8B aligned
#define BTPAD 34  // transposed B-tile row stride (b32 scatter, only 4B align needed)
#define KPAD 68   // K-tile row stride: 272B rows (16B-aligned b128 writes)
#define VTPAD 18  // transposed V-tile row stride

// ---------------------------------------------------------------------------
// CDNA5 async copies: global -> LDS, tracked by ASYNCcnt.
// LDS offset = low 32 bits of the generic pointer.
// ---------------------------------------------------------------------------
__device__ __forceinline__ unsigned lds_off(const void* p) {
  return (unsigned)(size_t)p;
}
__device__ __forceinline__ void async_ld_b128(unsigned ldsoff, const float* g) {
  asm volatile("global_load_async_to_lds_b128 %0, %1, off"
               :: "v"(ldsoff), "v"(g)
               : "memory");
}
__device__ __forceinline__ void async_ld_b32(unsigned ldsoff, const float* g) {
  asm volatile("global_load_async_to_lds_b32 %0, %1, off"
               :: "v"(ldsoff), "v"(g)
               : "memory");
}

// ---------------------------------------------------------------------------
// Kernel 1/4: fp32 WMMA GEMM  C[M,N] = A[M,K] * B[K,N]
// 128 threads (4 wave32 waves), 64x64 tile, K-step 32, double-buffered
// async-to-LDS staging. A staged row-major via b128 asyncs; B staged
// TRANSPOSED via per-lane b32 asyncs so both fragments are contiguous
// 8B LDS reads feeding V_WMMA_F32_16X16X4_F32 with no operand repacking.
// ---------------------------------------------------------------------------
__global__ __launch_bounds__(128) void gemm_f32_wmma(
    const float* __restrict__ A, const float* __restrict__ B,
    float* __restrict__ C, int M, int N, int Kdim, int lda, int ldb, int ldc) {
  __shared__ float As[2][64][APAD];    // row-major A tile [m][k]
  __shared__ float Bs[2][64][BTPAD];   // transposed B tile [n][k]

  const int tid = threadIdx.x;
  const int w = tid >> 5;
  const int lane = tid & 31;
  const int half_id = lane >> 4;
  const int lm = lane & 15;
  const int m0 = blockIdx.y * 64;
  const int n0 = blockIdx.x * 64;

  v8f acc[4] = {{}, {}, {}, {}};

  auto stage = [&](int buf, int k0) {
    // A: 64 rows x 32 floats = 512 float4 -> 4 b128 asyncs/thread
#pragma unroll
    for (int i = 0; i < 4; ++i) {
      int flat = tid + i * 128;
      int row = flat >> 3;
      int c4 = (flat & 7) * 4;
      async_ld_b128(lds_off(&As[buf][row][c4]),
                    A + (size_t)(m0 + row) * lda + k0 + c4);
    }
    // B transposed: 2048 dwords -> 16 b32 asyncs/thread.
    // flat = tid + i*128: n = flat&63 (fixed per thread), k = (tid>>6) + 2i.
    // Coalesced 128B/wave global reads; conflict-free stride-34 LDS scatter.
#pragma unroll
    for (int i = 0; i < 16; ++i) {
      int flat = tid + i * 128;
      int k = flat >> 6;
      int n = flat & 63;
      async_ld_b32(lds_off(&Bs[buf][n][k]),
                   B + (size_t)(k0 + k) * ldb + n0 + n);
    }
  };

  const int nk = Kdim / 32;
  stage(0, 0);

  for (int t0 = 0; t0 < nk; ++t0) {
    const int cur = t0 & 1;
    if (t0 + 1 < nk) {
      stage(cur ^ 1, (t0 + 1) * 32);
      asm volatile("s_wait_asynccnt 20" ::: "memory");  // current tile's 20 done
    } else {
      asm volatile("s_wait_asynccnt 0" ::: "memory");
    }
    __syncthreads();

    const int rb = w * 16;
#pragma unroll
    for (int kk = 0; kk < 8; ++kk) {
      const int c = kk * 4 + 2 * half_id;
      // A fragment: A[M=lm][K=c,c+1]  (contiguous 8B)
      v2f af = *(const v2f*)&As[cur][rb + lm][c];
#pragma unroll
      for (int t = 0; t < 4; ++t) {
        // B fragment: B[K=c,c+1][N=t*16+lm] from transposed [n][k] tile (contiguous 8B)
        v2f bf = *(const v2f*)&Bs[cur][t * 16 + lm][c];
        acc[t] = __builtin_amdgcn_wmma_f32_16x16x4_f32(
            false, af, false, bf, (short)0, acc[t], false, false);
      }
    }
    __syncthreads();   // all waves done reading before buffer is overwritten
  }

  // Epilogue: C layout row = r + 8*half_id, col = t*16 + lm
#pragma unroll
  for (int t = 0; t < 4; ++t) {
#pragma unroll
    for (int r = 0; r < 8; ++r) {
      int row = m0 + w * 16 + r + 8 * half_id;
      C[(size_t)row * ldc + n0 + t * 16 + lm] = acc[t][r];
    }
  }
}

// ---------------------------------------------------------------------------
// Kernel 2/4: interleaved RoPE applied in-place to q and k inside qkv buffer.
// ---------------------------------------------------------------------------
__global__ __launch_bounds__(256) void rope_kernel(
    float* __restrict__ qkv, const float* __restrict__ cosb,
    const float* __restrict__ sinb) {
  int idx = blockIdx.x * blockDim.x + threadIdx.x;
  int pair = idx & 31;
  int h = (idx >> 5) & 15;
  int qk = (idx >> 9) & 1;
  int bs = idx >> 10;                  // 0..4095
  int s = bs & (SS - 1);
  size_t off = (size_t)bs * LDQKV + qk * DD + h * DH + pair * 2;
  float c = cosb[s * DH + pair * 2];   // repeat(...,2): even==odd entry
  float sn = sinb[s * DH + pair * 2];
  float x0 = qkv[off], x1 = qkv[off + 1];
  qkv[off]     = x0 * c - x1 * sn;
  qkv[off + 1] = x1 * c + x0 * sn;
}

// ---------------------------------------------------------------------------
// Kernel 3/4: dual-triangle flash attention.
// Grid: (S/64, B*H), block 128 (4 waves). K tile staged row-major (b128
// asyncs), V tile staged TRANSPOSED [dim][key] (b32 asyncs) so P@V B
// fragments are contiguous 8B LDS reads. Both double-buffered across key
// blocks and shared by all 4 waves. Streaming softmax; P re-layouted C->A
// via wave-private LDS.
// ---------------------------------------------------------------------------
__global__ __launch_bounds__(128) void flash_attn_kernel(
    const float* __restrict__ qkv, float* __restrict__ attn_out) {
  __shared__ float Ks[2][16][KPAD];    // [key][dim]
  __shared__ float Vt[2][64][VTPAD];   // [dim][key]
  __shared__ float Plds[4][16][18];

  const int tid = threadIdx.x;
  const int w = tid >> 5;
  const int lane = tid & 31;
  const int half_id = lane >> 4;
  const int lm = lane & 15;
  const int bh = blockIdx.y;
  const int b = bh >> 4;
  const int h = bh & 15;
  const int q0 = blockIdx.x * 64 + w * 16;

  const size_t base = (size_t)b * SS * LDQKV + (size_t)h * DH;
  const float* Q = qkv + base;
  const float* K = qkv + base + DD;
  const float* V = qkv + base + 2 * DD;

  auto stageKV = [&](int buf, int j0) {
    // K tile: 16 rows x 64 floats = 256 float4 -> 2 b128 asyncs/thread
#pragma unroll
    for (int i = 0; i < 2; ++i) {
      int flat = tid + i * 128;        // 0..255
      int row = flat >> 4;             // key row 0..15
      int c4 = (flat & 15) * 4;        // dim 0..60
      async_ld_b128(lds_off(&Ks[buf][row][c4]),
                    K + (size_t)(j0 + row) * LDQKV + c4);
    }
    // V tile transposed: 1024 dwords -> 8 b32 asyncs/thread.
    // flat = tid + i*128: dim n = flat&63, key k = (tid>>6) + 2i.
#pragma unroll
    for (int i = 0; i < 8; ++i) {
      int flat = tid + i * 128;
      int k = flat >> 6;
      int n = flat & 63;
      async_ld_b32(lds_off(&Vt[buf][n][k]),
                   V + (size_t)(j0 + k) * LDQKV + n);
    }
  };

  // Preload Q fragments for both halves (loop-invariant)
  v2f qu[8], qd[8];
  {
    const float* qrow = Q + (size_t)(q0 + lm) * LDQKV;
#pragma unroll
    for (int kk = 0; kk < 8; ++kk) {
      int c = kk * 4 + 2 * half_id;
      qu[kk] = *(const v2f*)(qrow + c);        // dims 0..31
      qd[kk] = *(const v2f*)(qrow + 32 + c);   // dims 32..63
    }
  }

  float m_i[8], l_i[8];
  v8f O[4];
#pragma unroll
  for (int r = 0; r < 8; ++r) { m_i[r] = -3.0e38f; l_i[r] = 0.0f; }
#pragma unroll
  for (int t = 0; t < 4; ++t) {
#pragma unroll
    for (int r = 0; r < 8; ++r) O[t][r] = 0.0f;
  }

  const float scale = 0.17677669529663687f;    // 32^-0.5
  const int nblk = SS / 16;                    // 128

  stageKV(0, 0);

  for (int it = 0; it < nblk; ++it) {
    const int j0 = it * 16;
    const int cur = it & 1;
    if (it + 1 < nblk) {
      stageKV(cur ^ 1, j0 + 16);
      asm volatile("s_wait_asynccnt 10" ::: "memory");  // current K/V tile done
    } else {
      asm volatile("s_wait_asynccnt 0" ::: "memory");
    }
    __syncthreads();

    // S_up / S_down = Q_half @ K_half^T from LDS-resident K tile
    v8f su = {}, sd = {};
#pragma unroll
    for (int kk = 0; kk < 8; ++kk) {
      int c = kk * 4 + 2 * half_id;
      v2f ku = *(const v2f*)&Ks[cur][lm][c];        // B[K=c,c+1][N=lm]
      v2f kd = *(const v2f*)&Ks[cur][lm][32 + c];
      su = __builtin_amdgcn_wmma_f32_16x16x4_f32(
          false, qu[kk], false, ku, (short)0, su, false, false);
      sd = __builtin_amdgcn_wmma_f32_16x16x4_f32(
          false, qd[kk], false, kd, (short)0, sd, false, false);
    }

    // Dual-triangle select + online softmax (rows live in 16-lane halves)
    float p[8];
#pragma unroll
    for (int r = 0; r < 8; ++r) {
      int ig = q0 + r + 8 * half_id;
      int jg = j0 + lm;
      float s = (ig >= jg ? sd[r] : su[r]) * scale;
      float bm = s;
      bm = fmaxf(bm, __shfl_xor(bm, 1, 32));
      bm = fmaxf(bm, __shfl_xor(bm, 2, 32));
      bm = fmaxf(bm, __shfl_xor(bm, 4, 32));
      bm = fmaxf(bm, __shfl_xor(bm, 8, 32));
      float mn = fmaxf(m_i[r], bm);
      float pv = __expf(s - mn);
      float ps = pv;
      ps += __shfl_xor(ps, 1, 32);
      ps += __shfl_xor(ps, 2, 32);
      ps += __shfl_xor(ps, 4, 32);
      ps += __shfl_xor(ps, 8, 32);
      float alpha = __expf(m_i[r] - mn);
      l_i[r] = l_i[r] * alpha + ps;
      m_i[r] = mn;
      p[r] = pv;
#pragma unroll
      for (int t = 0; t < 4; ++t) O[t][r] *= alpha;
    }

    // P (C layout) -> A fragment layout via wave-private LDS tile
#pragma unroll
    for (int r = 0; r < 8; ++r) Plds[w][r + 8 * half_id][lm] = p[r];
    asm volatile("s_wait_dscnt 0" ::: "memory");  // DS in-order per wave; belt & braces

    // O += P @ V from transposed LDS-resident V tile
#pragma unroll
    for (int kk = 0; kk < 4; ++kk) {
      int c = kk * 4 + 2 * half_id;
      v2f af = *(const v2f*)&Plds[w][lm][c];       // A[M=lm][K=c,c+1]
#pragma unroll
      for (int t = 0; t < 4; ++t) {
        // B fragment: B[K=c,c+1][N=t*16+lm] = Vt[n][c..c+1]  (contiguous 8B)
        v2f bf = *(const v2f*)&Vt[cur][t * 16 + lm][c];
        O[t] = __builtin_amdgcn_wmma_f32_16x16x4_f32(
            false, af, false, bf, (short)0, O[t], false, false);
      }
    }
    __syncthreads();   // done reading K/V buffer before it is overwritten
  }

  // Normalize and write attn_out[b][s][h*64+d]
  const size_t orow0 = (size_t)b * SS * DD + (size_t)h * DH;
#pragma unroll
  for (int r = 0; r < 8; ++r) {
    float inv = 1.0f / l_i[r];
    int row = q0 + r + 8 * half_id;
#pragma unroll
    for (int t = 0; t < 4; ++t)
      attn_out[orow0 + (size_t)row * DD + t * 16 + lm] = O[t][r] * inv;
  }
}

// ---------------------------------------------------------------------------
// Host-side launch: qkv GEMM -> RoPE -> flash attention -> projection GEMM.
// Workspace: qkv (4096x3072 f32 = 48MB) + attn_out (4096x1024 f32 = 16MB).
// ---------------------------------------------------------------------------
extern "C" void kernel_launch(void* const* d_in, const int* in_sizes, int n_in,
                              void* d_out, int out_size, void* d_ws, size_t ws_size,
                              hipStream_t stream) {
  (void)in_sizes; (void)n_in; (void)out_size; (void)ws_size;
  const float* x        = (const float*)d_in[0];
  const float* w_qkv    = (const float*)d_in[1];
  const float* w_proj   = (const float*)d_in[2];
  const float* rope_cos = (const float*)d_in[3];
  const float* rope_sin = (const float*)d_in[4];
  float* out = (float*)d_out;

  float* qkv  = (float*)d_ws;                       // 4096 x 3072
  float* attn = qkv + (size_t)BB * SS * LDQKV;      // 4096 x 1024

  const dim3 blk(128);
  // qkv = x @ w_qkv   (M=4096, N=3072, K=1024)
  gemm_f32_wmma<<<dim3(3072 / 64, 4096 / 64), blk, 0, stream>>>(
      x, w_qkv, qkv, BB * SS, 3 * DD, DD, DD, 3 * DD, 3 * DD);
  // RoPE on q and k in place
  rope_kernel<<<(BB * SS * 2 * HH * 32) / 256, 256, 0, stream>>>(
      qkv, rope_cos, rope_sin);
  // dual-triangle flash attention
  flash_attn_kernel<<<dim3(SS / 64, BB * HH), blk, 0, stream>>>(qkv, attn);
  // out = attn @ w_proj   (M=4096, N=1024, K=1024)
  gemm_f32_wmma<<<dim3(1024 / 64, 4096 / 64), blk, 0, stream>>>(
      attn, w_proj, out, BB * SS, DD, DD, DD, DD, DD);
}